// LiquidGenerator_57724360458412
// MI455X (gfx1250) — compile-verified
//
#include <hip/hip_runtime.h>

typedef __attribute__((ext_vector_type(2))) float v2f;
typedef __attribute__((ext_vector_type(8))) float v8f;

#define CUTOFF_F 3.0f
#define EPS_F 1e-16f

// d_ws layout: float4 P[1024] (x,y,z,|p|^2), then float4 vec[27]

__global__ void liquid_gen_kernel(const float* __restrict__ pos,
                                  const float* __restrict__ tra,
                                  const float* __restrict__ rot,
                                  const float* __restrict__ cell,
                                  float4* __restrict__ Pws,
                                  float4* __restrict__ Vws,
                                  float* __restrict__ out) {
    int idx = blockIdx.x * blockDim.x + threadIdx.x;  // 0..1023
    if (idx >= 1024) return;
    if (idx == 0) *out = 0.0f;  // stream-ordered before score kernel

    int m = idx >> 3;

    float c0 = cell[0], c1 = cell[1], c2 = cell[2];
    float c3 = cell[3], c4 = cell[4], c5 = cell[5];
    float c6 = cell[6], c7 = cell[7], c8 = cell[8];

    float al = rot[3 * m + 0], be = rot[3 * m + 1], ga = rot[3 * m + 2];
    float ca = cosf(al), sa = sinf(al);
    float cb = cosf(be), sb = sinf(be);
    float cg = cosf(ga), sg = sinf(ga);

    // R = Rx @ Ry @ Rz
    float R00 = cb * cg,               R01 = -cb * sg,              R02 = -sb;
    float R10 = ca * sg - sa * sb * cg, R11 = ca * cg + sa * sb * sg, R12 = -sa * cb;
    float R20 = sa * sg + ca * sb * cg, R21 = sa * cg - ca * sb * sg, R22 = ca * cb;

    const float* pp = pos + idx * 3;
    float px = pp[0], py = pp[1], pz = pp[2];
    // gen = pos-row @ R  (einsum 'mai,mij->maj')
    float gx = px * R00 + py * R10 + pz * R20;
    float gy = px * R01 + py * R11 + pz * R21;
    float gz = px * R02 + py * R12 + pz * R22;

    float tx = tra[3 * m + 0]; tx -= floorf(tx);
    float ty = tra[3 * m + 1]; ty -= floorf(ty);
    float tz = tra[3 * m + 2]; tz -= floorf(tz);
    float Tx = tx * c0 + ty * c3 + tz * c6;
    float Ty = tx * c1 + ty * c4 + tz * c7;
    float Tz = tx * c2 + ty * c5 + tz * c8;

    float X = gx + Tx, Y = gy + Ty, Z = gz + Tz;
    Pws[idx] = make_float4(X, Y, Z, X * X + Y * Y + Z * Z);

    if (idx < 27) {
        float ox = (float)(idx / 9) - 1.0f;
        float oy = (float)((idx / 3) % 3) - 1.0f;
        float oz = (float)(idx % 3) - 1.0f;
        Vws[idx] = make_float4(ox * c0 + oy * c3 + oz * c6,
                               ox * c1 + oy * c4 + oz * c7,
                               ox * c2 + oy * c5 + oz * c8, 0.0f);
    }
}

// One wave per (16x16 tile, shift). 64*64*27 = 110592 waves; 256 thr = 8 waves/block.
__global__ void __launch_bounds__(256)
liquid_score_kernel(const float4* __restrict__ P,
                    const float4* __restrict__ V,
                    float* __restrict__ out) {
    const int wave = blockIdx.x * (blockDim.x >> 5) + (threadIdx.x >> 5);
    const int lane = threadIdx.x & 31;
    const int s    = wave >> 12;      // shift index 0..26
    const int rem  = wave & 4095;
    const int i0   = (rem >> 6) << 4; // row tile base
    const int j0   = (rem & 63) << 4; // col tile base
    const int half = lane >> 4;
    const int r    = lane & 15;

    float4 vec = V[s];
    float4 pa  = P[i0 + r];           // A row data (+ |p|^2 in .w)
    float4 pb  = P[j0 + r];           // B column source
    float sx = pb.x + vec.x, sy = pb.y + vec.y, sz = pb.z + vec.z;
    float snorm = sx * sx + sy * sy + sz * sz;

    // A: 16x4 f32, lanes 0-15 hold K=0,1; lanes 16-31 hold K=2,3 (K=3 padded 0)
    v2f A, B;
    A[0] = half ? pa.z : pa.x;
    A[1] = half ? 0.0f : pa.y;
    // B: 4x16 f32, mirrored layout per lane half
    B[0] = half ? sz : sx;
    B[1] = half ? 0.0f : sy;

    v8f acc = {};
    // D[m][n] = dot(p_{i0+m}, s_{j0+n})
    acc = __builtin_amdgcn_wmma_f32_16x16x4_f32(
        /*neg_a=*/false, A, /*neg_b=*/false, B,
        /*c_mod=*/(short)0, acc, /*reuse_a=*/false, /*reuse_b=*/false);

    // D layout: lane half h, VGPR v -> M = 8*h + v, N = lane&15
    float partial = 0.0f;
#pragma unroll
    for (int v = 0; v < 8; ++v) {
        float pn = __shfl(pa.w, (half << 3) + v, 32);  // |p_{i0+8h+v}|^2
        float d2 = fmaxf(pn + snorm - 2.0f * acc[v], 0.0f) + EPS_F;
        float d  = sqrtf(d2);
        float t  = CUTOFF_F - d;
        partial += (d < CUTOFF_F) ? (t * t) : 0.0f;
    }

#pragma unroll
    for (int off = 16; off > 0; off >>= 1)
        partial += __shfl_down(partial, off, 32);
    if (lane == 0) atomicAdd(out, partial);
}

extern "C" void kernel_launch(void* const* d_in, const int* in_sizes, int n_in,
                              void* d_out, int out_size, void* d_ws, size_t ws_size,
                              hipStream_t stream) {
    const float* positions   = (const float*)d_in[0]; // 128*8*3
    const float* translation = (const float*)d_in[1]; // 128*3
    const float* rotation    = (const float*)d_in[2]; // 128*3
    const float* cell        = (const float*)d_in[3]; // 9
    float* out = (float*)d_out;

    float4* Pws = (float4*)d_ws;        // 1024 * 16 B
    float4* Vws = Pws + 1024;           // 27 * 16 B

    liquid_gen_kernel<<<4, 256, 0, stream>>>(positions, translation, rotation,
                                             cell, Pws, Vws, out);

    // 64 row-tiles * 64 col-tiles * 27 shifts = 110592 waves / 8 per block
    liquid_score_kernel<<<13824, 256, 0, stream>>>(Pws, Vws, out);
}